// AnaphoricityScorer_45767171506489
// MI455X (gfx1250) — compile-verified
//
#include <hip/hip_runtime.h>
#include <hip/hip_bf16.h>

// ---------------------------------------------------------------------------
// AnaphoricityScorer for MI455X (gfx1250, wave32, WMMA)
//
// M=25600 pairs, K=3136 features, N=128 hidden, exact f32 via
// V_WMMA_F32_16X16X4_F32. M-tile=64 per WG (4 accumulators/wave) amortizes
// the L2-resident W1 reads 4x vs a 16-row tile; double-buffered LDS A-tile
// (stride 68 floats -> conflict-free b64 reads, b128 stores) with one
// barrier per K-chunk.
// ---------------------------------------------------------------------------

typedef float v2f __attribute__((ext_vector_type(2)));
typedef float v8f __attribute__((ext_vector_type(8)));

#define N_MENTIONS 10000
#define BATCH      512
#define N_ANTS     50
#define EMB        1024
#define PW_EMB     64
#define IN_FEAT    3136               // 3*EMB + PW_EMB
#define HIDDEN     128
#define M_TOTAL    (BATCH * N_ANTS)   // 25600
#define TILE_M     64
#define SUB_M      4                  // 4 x 16-row WMMA sub-tiles
#define KCHUNK     64
#define NCHUNKS    (IN_FEAT / KCHUNK) // 49
#define N_TILES_M  (M_TOTAL / TILE_M) // 400
#define ASTRIDE    68                 // floats; conflict-free + 16B-aligned rows
#define LRELU      0.01f
#define EPS        1e-7f

// --- pre-kernel: transpose W1[3136][128] -> Wt[128][3136] -------------------
__global__ __launch_bounds__(256) void w1_transpose_kernel(
    const float* __restrict__ W1, float* __restrict__ Wt) {
  int i = blockIdx.x * 256 + threadIdx.x;
  if (i < IN_FEAT * HIDDEN) {
    int k = i / HIDDEN;
    int c = i % HIDDEN;
    Wt[c * IN_FEAT + k] = W1[i];
  }
}

// --- epsilon dummy column ---------------------------------------------------
__global__ void dummy_col_kernel(float* __restrict__ out) {
  int b = blockIdx.x * 256 + threadIdx.x;
  if (b < BATCH) out[b * (N_ANTS + 1)] = EPS;
}

// segment-select load of one float4 of pair features for chunk c
__device__ __forceinline__ float4 load_feat(int c, int q4,
                                            const float* __restrict__ aptr,
                                            const float* __restrict__ gptr,
                                            const float* __restrict__ pptr) {
  const int f = c * KCHUNK + q4 * 4;
  if (c < 16) {                                   // a_mentions
    return *reinterpret_cast<const float4*>(aptr + f);
  } else if (c < 32) {                            // gathered b_mentions
    return *reinterpret_cast<const float4*>(gptr + (f - EMB));
  } else if (c < 48) {                            // similarity a*b
    float4 x = *reinterpret_cast<const float4*>(aptr + (f - 2 * EMB));
    float4 y = *reinterpret_cast<const float4*>(gptr + (f - 2 * EMB));
    float4 r; r.x = x.x * y.x; r.y = x.y * y.y; r.z = x.z * y.z; r.w = x.w * y.w;
    return r;
  } else {                                        // pw features
    return *reinterpret_cast<const float4*>(pptr + (f - 3 * EMB));
  }
}

// --- main fused kernel ------------------------------------------------------
__global__ __launch_bounds__(256) void anaphoricity_scorer_kernel(
    const float* __restrict__ all_mentions,   // [10000,1024]
    const float* __restrict__ mentions_batch, // [512,1024]
    const float* __restrict__ pw_batch,       // [512,50,64]
    const int*   __restrict__ top_indices,    // [512,50]
    const float* __restrict__ rough,          // [512,50]
    const float* __restrict__ Wt,             // [128,3136] transposed W1
    const float* __restrict__ b1,             // [128]
    const float* __restrict__ W_out,          // [128]
    const float* __restrict__ b_out,          // [1]
    float*       __restrict__ out)            // [512,51]
{
  __shared__ float As[2][TILE_M][ASTRIDE];  // double-buffered A tile
  __shared__ float red[TILE_M];

  const int t    = threadIdx.x;
  const int tile = blockIdx.x;
  const int P0   = tile * TILE_M;

  // ---- loader mapping: one pair-row per thread group of 4
  const int lrow = t >> 2;            // 0..63
  const int lq   = t & 3;             // float4 slot base
  const int p    = P0 + lrow;
  const int bb   = p / N_ANTS;
  const int idx  = top_indices[p];
  const float* aptr = mentions_batch + (size_t)bb  * EMB;
  const float* gptr = all_mentions   + (size_t)idx * EMB;
  const float* pptr = pw_batch       + (size_t)p   * PW_EMB;

  // ---- WMMA mapping: wave w owns hidden columns [16w, 16w+16)
  const int lane = t & 31;
  const int wave = t >> 5;
  const int n    = lane & 15;          // A row (within sub-tile) / B column
  const int hi   = lane >> 4;          // 0: K{0,1}, 1: K{2,3}
  const int koff = hi * 2;
  const int col  = wave * 16 + n;
  const float* wcol = Wt + (size_t)col * IN_FEAT + koff;

  v8f acc[SUB_M] = {};
  if (t < TILE_M) red[t] = 0.0f;

  // prologue: stage chunk 0
  {
    float4 v0 = load_feat(0, lq + 0,  aptr, gptr, pptr);
    float4 v1 = load_feat(0, lq + 4,  aptr, gptr, pptr);
    float4 v2 = load_feat(0, lq + 8,  aptr, gptr, pptr);
    float4 v3 = load_feat(0, lq + 12, aptr, gptr, pptr);
    *reinterpret_cast<float4*>(&As[0][lrow][(lq + 0)  * 4]) = v0;
    *reinterpret_cast<float4*>(&As[0][lrow][(lq + 4)  * 4]) = v1;
    *reinterpret_cast<float4*>(&As[0][lrow][(lq + 8)  * 4]) = v2;
    *reinterpret_cast<float4*>(&As[0][lrow][(lq + 12) * 4]) = v3;
  }
  __syncthreads();

  for (int c = 0; c < NCHUNKS; ++c) {
    const int cur = c & 1;
    const bool more = (c + 1) < NCHUNKS;

    // prefetch next chunk from global (overlaps this chunk's WMMAs)
    float4 v0, v1, v2, v3;
    if (more) {
      v0 = load_feat(c + 1, lq + 0,  aptr, gptr, pptr);
      v1 = load_feat(c + 1, lq + 4,  aptr, gptr, pptr);
      v2 = load_feat(c + 1, lq + 8,  aptr, gptr, pptr);
      v3 = load_feat(c + 1, lq + 12, aptr, gptr, pptr);
    }

    // 64 WMMAs: 16 K-steps x 4 M sub-tiles, B fragment reused across sub-tiles
    const float* wk = wcol + c * KCHUNK;
#pragma unroll
    for (int ks = 0; ks < KCHUNK / 4; ++ks) {
      v2f b = *reinterpret_cast<const v2f*>(wk + ks * 4);   // global_load_b64
#pragma unroll
      for (int m = 0; m < SUB_M; ++m) {
        v2f a = *reinterpret_cast<const v2f*>(
            &As[cur][m * 16 + n][ks * 4 + koff]);           // ds_load_b64
        acc[m] = __builtin_amdgcn_wmma_f32_16x16x4_f32(
            false, a, false, b, (short)0, acc[m], false, false);
      }
    }

    // stage next chunk into the other buffer, single barrier per chunk
    if (more) {
      *reinterpret_cast<float4*>(&As[cur ^ 1][lrow][(lq + 0)  * 4]) = v0;
      *reinterpret_cast<float4*>(&As[cur ^ 1][lrow][(lq + 4)  * 4]) = v1;
      *reinterpret_cast<float4*>(&As[cur ^ 1][lrow][(lq + 8)  * 4]) = v2;
      *reinterpret_cast<float4*>(&As[cur ^ 1][lrow][(lq + 12) * 4]) = v3;
    }
    __syncthreads();
  }

  // ---- epilogue: bias + LeakyReLU + W_out, reduce over 128 hidden units
  const float bias = b1[col];
  const float wo   = W_out[col];
#pragma unroll
  for (int m = 0; m < SUB_M; ++m) {
#pragma unroll
    for (int v = 0; v < 8; ++v) {
      float x = acc[m][v] + bias;       // C layout: VGPR v, lanes0-15 = row v,
      x = x > 0.0f ? x : LRELU * x;     //            lanes16-31 = row v+8
      x *= wo;
      for (int off = 1; off < 16; off <<= 1) x += __shfl_xor(x, off, 32);
      if (n == 0) atomicAdd(&red[m * 16 + v + hi * 8], x);
    }
  }
  __syncthreads();

  if (t < TILE_M) {
    int pp = P0 + t;
    int ob = pp / N_ANTS;
    int oa = pp % N_ANTS;
    out[ob * (N_ANTS + 1) + 1 + oa] = rough[pp] + red[t] + b_out[0];
  }
}

// ---------------------------------------------------------------------------
extern "C" void kernel_launch(void* const* d_in, const int* in_sizes, int n_in,
                              void* d_out, int out_size, void* d_ws, size_t ws_size,
                              hipStream_t stream) {
  const float* all_mentions = (const float*)d_in[0];
  const float* mentions     = (const float*)d_in[1];
  const float* pw           = (const float*)d_in[2];
  const int*   top_idx      = (const int*)  d_in[3];
  const float* rough        = (const float*)d_in[4];
  const float* W1           = (const float*)d_in[5];
  const float* b1           = (const float*)d_in[6];
  const float* W_out        = (const float*)d_in[7];
  const float* b_out        = (const float*)d_in[8];
  float*       out          = (float*)d_out;
  float*       Wt           = (float*)d_ws;   // 128*3136*4 = 1.6 MB scratch

  w1_transpose_kernel<<<(IN_FEAT * HIDDEN + 255) / 256, 256, 0, stream>>>(W1, Wt);
  dummy_col_kernel<<<(BATCH + 255) / 256, 256, 0, stream>>>(out);
  anaphoricity_scorer_kernel<<<N_TILES_M, 256, 0, stream>>>(
      all_mentions, mentions, pw, top_idx, rough, Wt, b1, W_out, b_out, out);
}